// _InferLevel_15891378995270
// MI455X (gfx1250) — compile-verified
//
#include <hip/hip_runtime.h>
#include <hip/hip_bf16.h>
#include <stdint.h>

// Trilinear hashed-voxel feature interpolation, MI455X (gfx1250, wave32).
// Memory-bound gather kernel: WMMA does not apply (per-point batched GEMV
// with per-point gathered B). CDNA5 features used:
//   * global_load_async_to_lds_b128 (+ s_wait_asynccnt) double-buffered gathers
//   * non-temporal streaming loads/stores so the 128 MB feature table + 8 MB
//     hash table stay resident in the 192 MB L2 while the 64 MB output streams
//   * v_readlane_b32 SALU broadcasts to keep LDS banks free for the data path

#define NPTS     (1 << 19)
#define DFEAT    32
#define NBUCKETS (1 << 21)
#define NGROUPS  (NPTS / 4)   // 4 points per wave-iteration

__device__ __forceinline__ void async_gather_b128(unsigned lds_off, unsigned voff,
                                                  const void* sbase) {
    // GVS form: mem_addr = SADDR + VADDR32 + inst_offset; vdst = LDS byte offset
    asm volatile("global_load_async_to_lds_b128 %0, %1, %2"
                 :: "v"(lds_off), "v"(voff), "s"(sbase)
                 : "memory");
}

__device__ __forceinline__ float readlane_f32(float v, int srclane) {
    return __int_as_float(__builtin_amdgcn_readlane(__float_as_int(v), srclane));
}

__global__ void __launch_bounds__(256)
trilerp_hash_kernel(const float* __restrict__ pts,
                    const float* __restrict__ vfeat,
                    const int*   __restrict__ hash2vox,
                    const float* __restrict__ smin,
                    float*       __restrict__ out)
{
    __shared__ __align__(16) float smem[8 * 2 * 32 * 32];   // 8 waves x 2 bufs x 4 KB = 64 KB

    const int lane  = threadIdx.x & 31;
    const int wid   = threadIdx.x >> 5;                       // wave in block (0..7)
    const int gwave = blockIdx.x * (blockDim.x >> 5) + wid;
    const int nwave = gridDim.x * (blockDim.x >> 5);

    const float s0 = smin[0], s1 = smin[1], s2 = smin[2];
    const float inv_res = 40.0f;                              // 1 / 0.025

    // Low 32 bits of a generic shared pointer == addrspace(3) byte offset,
    // which is what the async-LDS path's VDST VGPR expects.
    const unsigned lds_wave = (unsigned)(uintptr_t)(&smem[wid * 2048]);

    const int p_sub  = lane >> 3;   // point within group for meta phase (0..3)
    const int corner = lane & 7;    // corner for meta phase (0..7)
    const unsigned cx = corner & 1, cy = (corner >> 1) & 1, cz = (corner >> 2) & 1;

    // ---- per-(point,corner) hash / weight / voxel-id (one lane each) ----
    auto meta = [&](int g, float& w, int& vid) {
        const int pt = g * 4 + p_sub;
        const float px = __builtin_nontemporal_load(&pts[pt * 3 + 0]);
        const float py = __builtin_nontemporal_load(&pts[pt * 3 + 1]);
        const float pz = __builtin_nontemporal_load(&pts[pt * 3 + 2]);
        const float qx = (px - s0) * inv_res;
        const float qy = (py - s1) * inv_res;
        const float qz = (pz - s2) * inv_res;
        const float bx = floorf(qx), by = floorf(qy), bz = floorf(qz);
        const float fx = qx - bx, fy = qy - by, fz = qz - bz;
        const unsigned ix = (unsigned)(int)bx + cx;
        const unsigned iy = (unsigned)(int)by + cy;
        const unsigned iz = (unsigned)(int)bz + cz;
        const unsigned hv = (ix * 1u + iy * 2654435761u + iz * 805459861u) & (NBUCKETS - 1);
        const int v = hash2vox[hv];
        const float wx = cx ? fx : 1.0f - fx;
        const float wy = cy ? fy : 1.0f - fy;
        const float wz = cz ? fz : 1.0f - fz;
        w   = (v >= 0) ? wx * wy * wz : 0.0f;
        vid = (v >= 0) ? v : 0;
    };

    // ---- async-gather 32 feature rows (4 KB) of a group into LDS buffer ----
    const unsigned chunk = (unsigned)(lane & 7) * 16u;        // 16 B sub-chunk of a 128 B row
    const unsigned rsub  = (unsigned)(lane >> 3);             // row-within-quad (0..3)
    auto issue = [&](int buf, int vid) {
        const unsigned base = lds_wave + (unsigned)buf * 4096u;
#pragma unroll
        for (int k = 0; k < 8; ++k) {                         // 8 x b128 = 4 KB
            const int r  = k * 4 + (int)rsub;                 // gathered row 0..31
            const int vr = __shfl(vid, r, 32);                // per-lane index -> ds_bpermute
            const unsigned voff = (unsigned)vr * 128u + chunk;
            async_gather_b128(base + (unsigned)r * 128u + chunk, voff, vfeat);
        }
    };

    // ---- weighted accumulate from LDS, NT-stream the result ----
    auto compute = [&](int g, int buf, float w) {
        const float* b = &smem[wid * 2048 + buf * 1024];
#pragma unroll
        for (int p = 0; p < 4; ++p) {
            float acc = 0.0f;
#pragma unroll
            for (int c = 0; c < 8; ++c) {
                const int r = p * 8 + c;                      // compile-time lane index
                const float wr = readlane_f32(w, r);          // v_readlane_b32 (SALU bcast)
                acc = fmaf(wr, b[r * 32 + lane], acc);
            }
            __builtin_nontemporal_store(acc, &out[(size_t)(g * 4 + p) * DFEAT + lane]);
        }
    };

    // ---- software pipeline: double-buffered async gathers ----
    int g = gwave;
    if (g >= NGROUPS) return;

    float w0; int v0;
    meta(g, w0, v0);
    issue(0, v0);

    int buf = 0;
    while (g < NGROUPS) {
        const int gn = g + nwave;
        float w1 = 0.0f; int v1 = 0;
        if (gn < NGROUPS) {
            meta(gn, w1, v1);
            issue(buf ^ 1, v1);
            asm volatile("s_wait_asynccnt 0x8" ::: "memory");  // prev group's 8 loads done
        } else {
            asm volatile("s_wait_asynccnt 0x0" ::: "memory");
        }
        compute(g, buf, w0);
        w0 = w1; v0 = v1;
        g = gn; buf ^= 1;
    }
}

extern "C" void kernel_launch(void* const* d_in, const int* in_sizes, int n_in,
                              void* d_out, int out_size, void* d_ws, size_t ws_size,
                              hipStream_t stream) {
    const float* pts      = (const float*)d_in[0];
    const float* vfeat    = (const float*)d_in[1];
    const int*   hash2vox = (const int*)d_in[2];
    const float* smin     = (const float*)d_in[3];
    float*       out      = (float*)d_out;
    (void)in_sizes; (void)n_in; (void)out_size; (void)d_ws; (void)ws_size;

    // 2048 blocks x 256 threads = 16384 waves; 131072 groups -> 8 groups/wave.
    trilerp_hash_kernel<<<2048, 256, 0, stream>>>(pts, vfeat, hash2vox, smin, out);
}